// BlockAttentionResidual_88407606820975
// MI455X (gfx1250) — compile-verified
//
#include <hip/hip_runtime.h>

typedef __attribute__((ext_vector_type(2))) float v2f;
typedef __attribute__((ext_vector_type(8))) float v8f;

#define NBLK 9            // 8 prior blocks + current x
#define BB   4
#define CC   256
#define HW   4096         // 64*64
#define CHW  (CC * HW)    // 1,048,576 elements per (n,b) sample
#define GN_EPS 1e-5f

// ---------------------------------------------------------------------------
// Kernel 0: zero the (n,b) sum / sumsq accumulators (d_ws is poisoned 0xAA)
// ---------------------------------------------------------------------------
__global__ void BAR_zero_stats(float* __restrict__ stats) {
    int t = threadIdx.x;
    if (t < NBLK * BB * 2) stats[t] = 0.0f;
}

// ---------------------------------------------------------------------------
// Kernel 1: streaming pass over V.
//   - per-(n,b) sum & sum of squares (GroupNorm stats) via wave32 shuffle +
//     LDS + global atomicAdd
//   - channel dot  dot[n,b,hw] = sum_c (w*gamma)[c] * V[n,b,c,hw]
//     computed with V_WMMA_F32_16X16X4_F32: wave owns 16 hw columns (N dim),
//     A = broadcast fused weights (M dim wasted, but we are HBM-bound),
//     K = 4 channels per WMMA accumulated over the C loop.
// Operand striping (32-bit A/B, wave32):
//   lanes 0-15  : K = {0,1} in VGPR0/VGPR1
//   lanes 16-31 : K = {2,3} in VGPR0/VGPR1
// C/D: VGPR0 lanes 0-15 = row M=0, N=lane -> that is the dot result.
// ---------------------------------------------------------------------------
__global__ __launch_bounds__(256) void BAR_pass1(
    const float* __restrict__ blocks, const float* __restrict__ x,
    const float* __restrict__ w,      const float* __restrict__ gnw,
    float* __restrict__ stats,        float* __restrict__ dot)
{
    __shared__ float wg[CC];
    __shared__ float red[2];

    const int tid = threadIdx.x;
    wg[tid] = w[tid] * gnw[tid];          // fused w * gamma
    if (tid < 2) red[tid] = 0.0f;
    __syncthreads();

    const int chunks = HW / 128;          // 32 hw-chunks per (n,b)
    const int nb    = blockIdx.x / chunks;
    const int chunk = blockIdx.x % chunks;
    const int n = nb / BB;
    const int b = nb % BB;
    const float* base = (n < NBLK - 1)
        ? (blocks + (size_t)(n * BB + b) * CHW)
        : (x      + (size_t)b * CHW);

    const int wave = tid >> 5;
    const int lane = tid & 31;
    const int half = lane >> 4;           // 0: K={0,1}, 1: K={2,3}
    const int col  = lane & 15;           // N index (hw column)
    const int hw   = chunk * 128 + wave * 16 + col;

    const float* p = base + hw;
    v8f acc = {};                         // f32 16x16 accumulator (8 VGPRs)
    float s = 0.0f, ss = 0.0f;

    for (int c = 0; c < CC; c += 4) {
        const int r = c + 2 * half;
        const float v0 = p[(size_t)r * HW];
        const float v1 = p[(size_t)(r + 1) * HW];
        v2f av; av.x = wg[r];  av.y = wg[r + 1];
        v2f bv; bv.x = v0;     bv.y = v1;
        acc = __builtin_amdgcn_wmma_f32_16x16x4_f32(
                  false, av, false, bv, (short)0, acc, false, false);
        s  += v0 + v1;
        ss += v0 * v0 + v1 * v1;
    }

    if (lane < 16)                        // D row 0 = dot over all 256 channels
        dot[(size_t)nb * HW + hw] = acc[0];

    // wave32 butterfly reduction of stats, then LDS, then one global atomic
    for (int m = 16; m >= 1; m >>= 1) {
        s  += __shfl_xor(s,  m, 32);
        ss += __shfl_xor(ss, m, 32);
    }
    if (lane == 0) {
        atomicAdd(&red[0], s);
        atomicAdd(&red[1], ss);
    }
    __syncthreads();
    if (tid == 0) {
        atomicAdd(&stats[nb * 2 + 0], red[0]);
        atomicAdd(&stats[nb * 2 + 1], red[1]);
    }
}

// ---------------------------------------------------------------------------
// Kernel 2: fold stats into logits, softmax over the 9-deep axis.
//   logit[n] = istd[n,b] * (dot[n,b,hw] - mean[n,b]*S1) + S2
//   S1 = sum_c w*gamma,  S2 = sum_c w*beta
// One block = one b (4096 % 256 == 0), 64 blocks total.
// ---------------------------------------------------------------------------
__global__ __launch_bounds__(256) void BAR_pass2(
    const float* __restrict__ w,    const float* __restrict__ gnw,
    const float* __restrict__ gnb,  const float* __restrict__ stats,
    const float* __restrict__ dot,  float* __restrict__ attn)
{
    __shared__ float r1[256];
    __shared__ float r2[256];
    __shared__ float istd_s[NBLK], mean_s[NBLK];
    __shared__ float S12[2];

    const int tid = threadIdx.x;
    r1[tid] = w[tid] * gnw[tid];
    r2[tid] = w[tid] * gnb[tid];
    __syncthreads();
    for (int off = 128; off >= 1; off >>= 1) {
        if (tid < off) { r1[tid] += r1[tid + off]; r2[tid] += r2[tid + off]; }
        __syncthreads();
    }
    if (tid == 0) { S12[0] = r1[0]; S12[1] = r2[0]; }

    const int p0 = blockIdx.x * 256;
    const int b  = p0 / HW;               // uniform within the block
    if (tid < NBLK) {
        const float invM = 1.0f / (float)CHW;
        const float su = stats[(tid * BB + b) * 2 + 0];
        const float sq = stats[(tid * BB + b) * 2 + 1];
        const float mean = su * invM;
        const float var  = sq * invM - mean * mean;
        istd_s[tid] = rsqrtf(var + GN_EPS);
        mean_s[tid] = mean;
    }
    __syncthreads();

    const int hw = (p0 + tid) % HW;
    const float S1 = S12[0], S2 = S12[1];

    float lg[NBLK];
    float mx = -3.4e38f;
#pragma unroll
    for (int n = 0; n < NBLK; ++n) {
        const float d = dot[(size_t)(n * BB + b) * HW + hw];
        const float l = istd_s[n] * (d - mean_s[n] * S1) + S2;
        lg[n] = l;
        mx = fmaxf(mx, l);
    }
    float sum = 0.0f;
#pragma unroll
    for (int n = 0; n < NBLK; ++n) { lg[n] = __expf(lg[n] - mx); sum += lg[n]; }
    const float inv = 1.0f / sum;
#pragma unroll
    for (int n = 0; n < NBLK; ++n)
        attn[(size_t)(n * BB + b) * HW + hw] = lg[n] * inv;
}

// ---------------------------------------------------------------------------
// Kernel 3: h[b,c,hw] = sum_n attn[n,b,hw] * V[n,b,c,hw]
// float4 vectorized (global_load_b128 / global_store_b128); attn (576 KB)
// is reused by all 256 channels -> L2/L1 hits; V re-read should hit the
// 192 MB L2 since pass 1 streamed the identical 144 MB working set.
// ---------------------------------------------------------------------------
__global__ __launch_bounds__(256) void BAR_pass3(
    const float* __restrict__ blocks, const float* __restrict__ x,
    const float* __restrict__ attn,   float* __restrict__ out)
{
    const int g    = blockIdx.x * 256 + threadIdx.x;   // 0 .. B*C*HW/4 - 1
    const int hw4  = g & 1023;
    const int rest = g >> 10;
    const int c    = rest & 255;
    const int b    = rest >> 8;
    const int hw   = hw4 * 4;

    float4 acc = make_float4(0.0f, 0.0f, 0.0f, 0.0f);
#pragma unroll
    for (int n = 0; n < NBLK; ++n) {
        const float* base = (n < NBLK - 1)
            ? (blocks + (size_t)(n * BB + b) * CHW)
            : (x      + (size_t)b * CHW);
        const float4 v = *(const float4*)(base + (size_t)c * HW + hw);
        const float4 a = *(const float4*)(attn + (size_t)(n * BB + b) * HW + hw);
        acc.x += a.x * v.x;  acc.y += a.y * v.y;
        acc.z += a.z * v.z;  acc.w += a.w * v.w;
    }
    *(float4*)(out + (size_t)(b * CC + c) * HW + hw) = acc;
}

// ---------------------------------------------------------------------------
extern "C" void kernel_launch(void* const* d_in, const int* in_sizes, int n_in,
                              void* d_out, int out_size, void* d_ws, size_t ws_size,
                              hipStream_t stream) {
    const float* blocks = (const float*)d_in[0];   // [8,4,256,64,64]
    const float* x      = (const float*)d_in[1];   // [4,256,64,64]
    const float* w      = (const float*)d_in[2];   // [256]
    const float* gnw    = (const float*)d_in[3];   // [256]
    const float* gnb    = (const float*)d_in[4];   // [256]
    float* out = (float*)d_out;                    // [4,256,64,64]

    char* ws = (char*)d_ws;
    float* stats = (float*)ws;                                          // 72 floats
    float* dot   = (float*)(ws + 256);                                  // 9*4*4096
    float* attn  = (float*)(ws + 256 + (size_t)NBLK * BB * HW * 4);     // 9*4*4096

    BAR_zero_stats<<<1, 128, 0, stream>>>(stats);
    BAR_pass1<<<NBLK * BB * (HW / 128), 256, 0, stream>>>(blocks, x, w, gnw, stats, dot);
    BAR_pass2<<<(BB * HW) / 256, 256, 0, stream>>>(w, gnw, gnb, stats, dot, attn);
    BAR_pass3<<<(BB * CC * HW / 4) / 256, 256, 0, stream>>>(blocks, x, attn, out);
}